// Encoder_67121748902124
// MI455X (gfx1250) — compile-verified
//
#include <hip/hip_runtime.h>

// ---------------------------------------------------------------------------
// 3-layer GCN for MI455X (gfx1250, wave32).
//   per layer: Y = (X * norm_src) @ W  (f32 WMMA 16x16x4, dual M-tile)
//              agg = scatter_add(Y[src] -> dst)   (global_atomic_add_f32)
//              out = agg * norm_dst (+bias) (+relu)
// ---------------------------------------------------------------------------

typedef float v2f __attribute__((ext_vector_type(2)));
typedef float v4f __attribute__((ext_vector_type(4)));
typedef float v8f __attribute__((ext_vector_type(8)));

__device__ __forceinline__ void atomAddF(float* p, float v) {
  // Native global_atomic_add_f32 (agg buffer fits in 192MB L2 -> fast RMW).
  unsafeAtomicAdd(p, v);
}

// ------------------------- degree / norm ----------------------------------
__global__ void degree_kernel(const int* __restrict__ src, const int* __restrict__ dst,
                              float* __restrict__ degO, float* __restrict__ degI, int E) {
  int e = blockIdx.x * blockDim.x + threadIdx.x;
  if (e < E) {
    atomAddF(&degO[src[e]], 1.0f);
    atomAddF(&degI[dst[e]], 1.0f);
  }
}

__global__ void norm_kernel(float* __restrict__ a, float* __restrict__ b, int N) {
  int i = blockIdx.x * blockDim.x + threadIdx.x;
  if (i < N) {
    a[i] = rsqrtf(fmaxf(a[i], 1.0f));
    b[i] = rsqrtf(fmaxf(b[i], 1.0f));
  }
}

// ------------------------- scaled GEMM via WMMA ---------------------------
// Block = 32 rows of X (two 16-row M-tiles); one wave per 16-column tile.
// A staged in LDS (scaled by norm_src), stride FIN+4 => conflict-free
// wave32 b64 reads. Each B load pair feeds TWO independent WMMAs
// (separate accumulators -> no D->A/B hazard, back-to-back issue).
//
// V_WMMA_F32_16X16X4_F32 operand layout (ISA 7.12.2):
//   A (16x4): lane m=lane&15, kh=lane>>4 ; a[j] = A[m][k + 2*kh + j]
//   B (4x16): b[j] = B[k + 2*kh + j][n], n = lane&15
//   C/D     : vgpr r holds row (r + 8*kh), col n = lane&15
template <int FIN, int FOUT, int MT>
__global__ void gemm_scaled_kernel(const float* __restrict__ X, const float* __restrict__ nsrc,
                                   const float* __restrict__ W, float* __restrict__ Y, int N) {
  constexpr int NWAVES = FOUT / 16;
  constexpr int NT = NWAVES * 32;
  constexpr int LD = FIN + 4;
  constexpr int ROWS = MT * 16;
  __shared__ float Xs[ROWS * LD];

  const int row0 = blockIdx.x * ROWS;
  const int tid = threadIdx.x;

  // Stage scaled A tiles (coalesced); zero-pad OOB rows so WMMA runs unguarded.
  for (int i = tid; i < ROWS * FIN; i += NT) {
    int r = i / FIN;
    int c = i - r * FIN;
    int row = row0 + r;
    float v = 0.0f;
    if (row < N) v = X[(size_t)row * FIN + c] * nsrc[row];
    Xs[r * LD + c] = v;
  }
  __syncthreads();

  const int wv   = tid >> 5;
  const int ln   = tid & 31;
  const int m    = ln & 15;   // row (A) / col (B,D)
  const int kh   = ln >> 4;   // K half-select
  const int col0 = wv * 16;

  v8f acc[MT];
#pragma unroll
  for (int t = 0; t < MT; ++t) acc[t] = (v8f){};

#pragma unroll 4
  for (int k = 0; k < FIN; k += 4) {
    const int kk = k + 2 * kh;
    v2f b;
    b[0] = W[(size_t)kk * FOUT + col0 + m];
    b[1] = W[(size_t)(kk + 1) * FOUT + col0 + m];
#pragma unroll
    for (int t = 0; t < MT; ++t) {
      v2f a;
      a[0] = Xs[(t * 16 + m) * LD + kk];
      a[1] = Xs[(t * 16 + m) * LD + kk + 1];
      // (neg_a, A, neg_b, B, c_mod, C, reuse_a, reuse_b)
      acc[t] = __builtin_amdgcn_wmma_f32_16x16x4_f32(false, a, false, b, (short)0,
                                                     acc[t], false, false);
    }
  }

  const bool full = (row0 + ROWS) <= N;  // uniform per block
#pragma unroll
  for (int t = 0; t < MT; ++t) {
    const int rbase = row0 + t * 16 + 8 * kh;
    float* yp = Y + (size_t)rbase * FOUT + col0 + m;
    if (full) {
#pragma unroll
      for (int r = 0; r < 8; ++r) yp[(size_t)r * FOUT] = acc[t][r];
    } else {
#pragma unroll
      for (int r = 0; r < 8; ++r)
        if (rbase + r < N) yp[(size_t)r * FOUT] = acc[t][r];
    }
  }
}

// ------------------------- edge scatter-add -------------------------------
// One wave per edge: coalesced b128/b64 row read, native f32 atomics to dst.
template <int F>
__global__ void scatter_add_kernel(const float* __restrict__ H, const int* __restrict__ src,
                                   const int* __restrict__ dst, float* __restrict__ A, int E) {
  constexpr int VEC = F / 32;
  int gw = (int)((blockIdx.x * (size_t)blockDim.x + threadIdx.x) >> 5);
  int ln = threadIdx.x & 31;
  if (gw >= E) return;
  int s = src[gw];
  int d = dst[gw];
  const float* hp = H + (size_t)s * F + ln * VEC;
  float*       ap = A + (size_t)d * F + ln * VEC;
  if (VEC == 4) {
    v4f v = *(const v4f*)hp;
    atomAddF(ap + 0, v[0]);
    atomAddF(ap + 1, v[1]);
    atomAddF(ap + 2, v[2]);
    atomAddF(ap + 3, v[3]);
  } else {
    v2f v = *(const v2f*)hp;
    atomAddF(ap + 0, v[0]);
    atomAddF(ap + 1, v[1]);
  }
}

// ------------------------- finalize (scale / relu / bias) -----------------
template <int F, bool RELU, bool BIAS>
__global__ void finalize_kernel(const float* __restrict__ agg, const float* __restrict__ ndst,
                                const float* __restrict__ bias, float* __restrict__ out, int N) {
  int i = blockIdx.x * blockDim.x + threadIdx.x;
  if (i >= N * F) return;
  int row = i / F;
  int col = i - row * F;
  float v = agg[i] * ndst[row];
  if (BIAS) v += bias[col];
  if (RELU) v = fmaxf(v, 0.0f);
  out[i] = v;
}

// ---------------------------------------------------------------------------
static inline size_t alignUp(size_t x, size_t a) { return (x + a - 1) / a * a; }

extern "C" void kernel_launch(void* const* d_in, const int* in_sizes, int n_in,
                              void* d_out, int out_size, void* d_ws, size_t ws_size,
                              hipStream_t stream) {
  const float* x   = (const float*)d_in[0];
  const int*   src = (const int*)d_in[1];
  const int*   dst = (const int*)d_in[2];
  const float* W1  = (const float*)d_in[3];
  const float* W2  = (const float*)d_in[4];
  const float* W3  = (const float*)d_in[5];
  const float* b3  = (const float*)d_in[6];
  float* out = (float*)d_out;

  const int N = in_sizes[0] / 128;
  const int E = in_sizes[1];
  (void)n_in; (void)out_size; (void)ws_size;

  // Workspace: norms + two N x 128 ping-pong buffers (~103 MB).
  char* p = (char*)d_ws;
  float* nsrc = (float*)p; p += alignUp((size_t)N * sizeof(float), 256);
  float* ndst = (float*)p; p += alignUp((size_t)N * sizeof(float), 256);
  float* bufA = (float*)p; p += alignUp((size_t)N * 128 * sizeof(float), 256);
  float* bufB = (float*)p;

  const int mtiles        = (N + 31) / 32;   // 32 rows per block
  const int scatterBlocks = (E + 7) / 8;     // 8 waves (edges) per 256-thread block
  const int elemBlocks128 = (N * 128 + 255) / 256;
  const int elemBlocks64  = (N * 64 + 255) / 256;

  // ---- degrees -> norms ----
  hipMemsetAsync(nsrc, 0, (size_t)N * sizeof(float), stream);
  hipMemsetAsync(ndst, 0, (size_t)N * sizeof(float), stream);
  degree_kernel<<<(E + 255) / 256, 256, 0, stream>>>(src, dst, nsrc, ndst, E);
  norm_kernel<<<(N + 255) / 256, 256, 0, stream>>>(nsrc, ndst, N);

  // ---- layer 1: x -> bufA(gemm) -> bufB(agg) -> bufA(h1, relu) ----
  gemm_scaled_kernel<128, 128, 2><<<mtiles, 256, 0, stream>>>(x, nsrc, W1, bufA, N);
  hipMemsetAsync(bufB, 0, (size_t)N * 128 * sizeof(float), stream);
  scatter_add_kernel<128><<<scatterBlocks, 256, 0, stream>>>(bufA, src, dst, bufB, E);
  finalize_kernel<128, true, false><<<elemBlocks128, 256, 0, stream>>>(bufB, ndst, nullptr, bufA, N);

  // ---- layer 2: bufA -> bufB(gemm) -> bufA(agg) -> bufB(h2, relu) ----
  gemm_scaled_kernel<128, 128, 2><<<mtiles, 256, 0, stream>>>(bufA, nsrc, W2, bufB, N);
  hipMemsetAsync(bufA, 0, (size_t)N * 128 * sizeof(float), stream);
  scatter_add_kernel<128><<<scatterBlocks, 256, 0, stream>>>(bufB, src, dst, bufA, E);
  finalize_kernel<128, true, false><<<elemBlocks128, 256, 0, stream>>>(bufA, ndst, nullptr, bufB, N);

  // ---- layer 3 (Fout=64): bufB -> bufA(gemm) -> bufB(agg) -> out(+bias) ----
  gemm_scaled_kernel<128, 64, 2><<<mtiles, 128, 0, stream>>>(bufB, nsrc, W3, bufA, N);
  hipMemsetAsync(bufB, 0, (size_t)N * 64 * sizeof(float), stream);
  scatter_add_kernel<64><<<scatterBlocks, 256, 0, stream>>>(bufA, src, dst, bufB, E);
  finalize_kernel<64, false, true><<<elemBlocks64, 256, 0, stream>>>(bufB, ndst, b3, out, N);
}